// SimpleSparseConvNet_87033217286291
// MI455X (gfx1250) — compile-verified
//
#include <hip/hip_runtime.h>

typedef __attribute__((ext_vector_type(2))) float v2f;
typedef __attribute__((ext_vector_type(8))) float v8f;

// ---------------------------------------------------------------------------
// Zero the output (segment_sum accumulator). d_out is poisoned by the harness.
// ---------------------------------------------------------------------------
__global__ void zero_out_kernel(float* __restrict__ out, int n) {
    int i = blockIdx.x * blockDim.x + threadIdx.x;
    if (i < n) out[i] = 0.0f;
}

// ---------------------------------------------------------------------------
// Sparse conv gather-WMMA-scatter.
// Each wave handles 16 rulebook pairs -> one 16x16 (pairs x Cout) tile via
// four accumulating V_WMMA_F32_16X16X4_F32 ops:
//   D[m, n] = sum_k A[m, k] * B[k, n],  A = diag(feat[m]) chunked over K,
//   B rows = W[k_idx[m], :].
// Scatter with f32 global atomics (output rows alias across k offsets).
// ---------------------------------------------------------------------------
__global__ __launch_bounds__(256)
void sparse_conv_wmma_kernel(const float* __restrict__ feats,
                             const float* __restrict__ weight,
                             const int*   __restrict__ in_idx,
                             const int*   __restrict__ out_idx,
                             const int*   __restrict__ k_idx,
                             float*       __restrict__ out,
                             int M, int wlen) {
    // Stage the (tiny) 27x16 weight table in LDS.
    __shared__ float sW[512];
    int tid = threadIdx.x;
    for (int i = tid; i < wlen; i += blockDim.x) sW[i] = weight[i];
    __syncthreads();

    const int  wave    = tid >> 5;
    const int  lane    = tid & 31;
    const int  m_local = lane & 15;        // pair-in-tile this lane loads / col N
    const bool hi      = lane >= 16;

    long base = ((long)blockIdx.x * 8 + wave) * 16;   // first pair of this tile
    if (base >= (long)M) return;                      // wave-uniform exit

    // Lane l loads rulebook entry (base + l%16); tail pairs contribute 0.
    long gm    = base + m_local;
    bool valid = gm < (long)M;
    int  gi = valid ? in_idx[gm]  : 0;
    int  go = valid ? out_idx[gm] : 0;
    int  gk = valid ? k_idx[gm]   : 0;
    float f = valid ? feats[gi]   : 0.0f;

    // Accumulate D = sum over 4 K-chunks of (diag-chunk A) x (W-rows B).
    v8f acc = {0.f, 0.f, 0.f, 0.f, 0.f, 0.f, 0.f, 0.f};
    const int kkb = hi ? 2 : 0;            // K index held in VGPR0 for this lane
#pragma unroll
    for (int c = 0; c < 4; ++c) {
        // A (16x4, 2 VGPRs): nonzero only on the diagonal of chunk c.
        float a0 = (m_local == 4 * c + kkb)     ? f : 0.0f;
        float a1 = (m_local == 4 * c + kkb + 1) ? f : 0.0f;
        // B (4x16, 2 VGPRs): row kk = W[k_idx[pair 4c+kk], :], col = lane%16.
        int k0 = __shfl(gk, 4 * c + kkb,     32);
        int k1 = __shfl(gk, 4 * c + kkb + 1, 32);
        float b0 = sW[k0 * 16 + m_local];
        float b1 = sW[k1 * 16 + m_local];
        v2f A = {a0, a1};
        v2f B = {b0, b1};
        // 8 args: (neg_a, A, neg_b, B, c_mod, C, reuse_a, reuse_b)
        acc = __builtin_amdgcn_wmma_f32_16x16x4_f32(
            false, A, false, B, (short)0, acc, false, false);
    }

    // Scatter: lane l holds D[rows rbase..rbase+7, col lane%16] in acc[0..7].
    const int rbase = hi ? 8 : 0;
#pragma unroll
    for (int v = 0; v < 8; ++v) {
        int r = rbase + v;
        int o = __shfl(go, r, 32);         // out row of pair r (full-EXEC shuffle)
        if (base + r < (long)M) {
            unsafeAtomicAdd(&out[(long)o * 16 + m_local], acc[v]);
        }
    }
}

extern "C" void kernel_launch(void* const* d_in, const int* in_sizes, int n_in,
                              void* d_out, int out_size, void* d_ws, size_t ws_size,
                              hipStream_t stream) {
    const float* feats  = (const float*)d_in[0];
    const float* weight = (const float*)d_in[1];
    const int*   in_idx  = (const int*)d_in[2];
    const int*   out_idx = (const int*)d_in[3];
    const int*   k_idx   = (const int*)d_in[4];
    float* out = (float*)d_out;

    const int M    = in_sizes[2];   // rulebook length
    const int wlen = in_sizes[1];   // 27*16 weight elements

    // 1) zero the accumulator
    int zt = 256;
    int zb = (out_size + zt - 1) / zt;
    zero_out_kernel<<<zb, zt, 0, stream>>>(out, out_size);

    // 2) gather-WMMA-scatter: 16 pairs/wave, 8 waves/block -> 128 pairs/block
    long groups = ((long)M + 15) / 16;
    int  blocks = (int)((groups + 7) / 8);
    if (blocks < 1) blocks = 1;
    sparse_conv_wmma_kernel<<<blocks, 256, 0, stream>>>(
        feats, weight, in_idx, out_idx, k_idx, out, M, wlen);
}